// adder2d_65867618451686
// MI455X (gfx1250) — compile-verified
//
#include <hip/hip_runtime.h>

#define NTHREADS 256
#define CI 64
#define CO 64
#define XCOLS 30
#define XSTRIDE 68   // padded ci-stride per (row,col) cell: 68 % 64 == 4 -> spreads banks
#define COH 32       // output channels resident in LDS per pass

#if defined(__AMDGCN__) && __has_builtin(__builtin_amdgcn_global_load_async_to_lds_b32)
#define HAVE_ASYNC 1
#else
#define HAVE_ASYNC 0
#endif

__device__ __forceinline__ void g2l_f32(const float* g, float* l) {
#if HAVE_ASYNC
    __builtin_amdgcn_global_load_async_to_lds_b32(
        (__attribute__((address_space(1))) int*)(float*)g,
        (__attribute__((address_space(3))) int*)l,
        0, 0);
#else
    *l = *g;
#endif
}

__device__ __forceinline__ void wait_async_zero() {
#if HAVE_ASYNC
#if __has_builtin(__builtin_amdgcn_s_wait_asynccnt)
    __builtin_amdgcn_s_wait_asynccnt(0);
#else
    asm volatile("s_wait_asynccnt 0" ::: "memory");
#endif
#endif
}

__global__ __launch_bounds__(NTHREADS)
void adder2d_kernel(const float* __restrict__ x,
                    const float* __restrict__ Wt,
                    float* __restrict__ out) {
    // x tile: rows (oh-1..oh+1) x padded cols (0..29) x 64 ci, ci innermost
    __shared__ __align__(16) float xs[3 * XCOLS * XSTRIDE];
    // weights for 32 co: [co_local][k=kh*3+kw][ci], ci innermost
    __shared__ __align__(16) float ws[COH * 9 * CI];

    const int oh  = blockIdx.x;          // 0..27
    const int n   = blockIdx.y;          // 0..15
    const int tid = threadIdx.x;
    const int ow  = tid & 31;            // lane-varying; lanes 28..31 idle in compute
    const int cs  = tid >> 5;            // wave-uniform co_sub 0..7 -> LDS broadcast

    // ---- stage x tile (once per block) via async global->LDS gather ----
    for (int e = tid; e < 3 * XCOLS * CI; e += NTHREADS) {
        const int cell = e >> 6;                  // r*30 + c
        const int ci   = e & 63;
        const int r    = cell / XCOLS;
        const int c    = cell - r * XCOLS;
        const int h    = oh - 1 + r;
        const int w    = c - 1;
        float* dst = &xs[cell * XSTRIDE + ci];
        if ((unsigned)h < 28u && (unsigned)w < 28u) {
            g2l_f32(&x[((n * CI + ci) * 28 + h) * 28 + w], dst);
        } else {
            *dst = 0.0f;                          // zero-pad halo
        }
    }

    for (int half = 0; half < 2; ++half) {
        // ---- stage 32 co of weights, transposed to [co][k][ci] ----
        for (int e = tid; e < COH * 9 * CI; e += NTHREADS) {
            const int ci  = e & 63;
            const int t   = e >> 6;               // co_l*9 + k
            const int col = t / 9;
            const int k   = t - col * 9;
            g2l_f32(&Wt[((half * COH + col) * CI + ci) * 9 + k], &ws[e]);
        }
        wait_async_zero();
        __syncthreads();

        if (ow < 28) {
            float a0 = 0.f, a1 = 0.f, a2 = 0.f, a3 = 0.f;
#pragma unroll
            for (int kh = 0; kh < 3; ++kh) {
#pragma unroll
                for (int kw = 0; kw < 3; ++kw) {
                    const float4* xp  = (const float4*)&xs[(kh * XCOLS + ow + kw) * XSTRIDE];
                    const int k = kh * 3 + kw;
                    const float4* wp0 = (const float4*)&ws[((cs +  0) * 9 + k) * CI];
                    const float4* wp1 = (const float4*)&ws[((cs +  8) * 9 + k) * CI];
                    const float4* wp2 = (const float4*)&ws[((cs + 16) * 9 + k) * CI];
                    const float4* wp3 = (const float4*)&ws[((cs + 24) * 9 + k) * CI];
#pragma unroll 4
                    for (int q = 0; q < CI / 4; ++q) {
                        const float4 xv = xp[q];
                        float4 wv;
                        wv = wp0[q];
                        a0 += __builtin_fabsf(xv.x - wv.x) + __builtin_fabsf(xv.y - wv.y)
                            + __builtin_fabsf(xv.z - wv.z) + __builtin_fabsf(xv.w - wv.w);
                        wv = wp1[q];
                        a1 += __builtin_fabsf(xv.x - wv.x) + __builtin_fabsf(xv.y - wv.y)
                            + __builtin_fabsf(xv.z - wv.z) + __builtin_fabsf(xv.w - wv.w);
                        wv = wp2[q];
                        a2 += __builtin_fabsf(xv.x - wv.x) + __builtin_fabsf(xv.y - wv.y)
                            + __builtin_fabsf(xv.z - wv.z) + __builtin_fabsf(xv.w - wv.w);
                        wv = wp3[q];
                        a3 += __builtin_fabsf(xv.x - wv.x) + __builtin_fabsf(xv.y - wv.y)
                            + __builtin_fabsf(xv.z - wv.z) + __builtin_fabsf(xv.w - wv.w);
                    }
                }
            }
            const int ob = ((n * CO + half * COH + cs) * 28 + oh) * 28 + ow;
            out[ob          ] = -a0;
            out[ob +  8*784 ] = -a1;
            out[ob + 16*784 ] = -a2;
            out[ob + 24*784 ] = -a3;
        }
        __syncthreads();   // ws about to be overwritten (or kernel end)
    }
}

extern "C" void kernel_launch(void* const* d_in, const int* in_sizes, int n_in,
                              void* d_out, int out_size, void* d_ws, size_t ws_size,
                              hipStream_t stream) {
    const float* x  = (const float*)d_in[0];   // (16,64,28,28) f32
    const float* Wt = (const float*)d_in[1];   // (64,64,3,3)  f32
    float* out = (float*)d_out;                // (16,64,28,28) f32
    dim3 grid(28, 16);
    adder2d_kernel<<<grid, NTHREADS, 0, stream>>>(x, Wt, out);
}